// Self_Attention_23605140259323
// MI455X (gfx1250) — compile-verified
//
#include <hip/hip_runtime.h>
#include <stdint.h>

typedef __attribute__((ext_vector_type(2))) float v2f;
typedef __attribute__((ext_vector_type(8))) float v8f;
typedef __attribute__((ext_vector_type(4))) unsigned int v4u;
typedef __attribute__((ext_vector_type(8))) int v8i;
typedef __attribute__((ext_vector_type(4))) int v4i;

#define Bsz  8
#define Cch  64
#define Npix 4096              // H*W
#define BCN  (Bsz * Cch * Npix)
#define JT   (Npix / 16)       // 256 16-wide tiles along N
#define CHUNK 64               // columns staged per TDM transfer
#define NCH  (Npix / CHUNK)    // 64 chunks
#define GROW 65                // padded LDS row stride in floats (TDM pad)

// D = A(16x4) * B(4x16) + C, full f32 WMMA
__device__ __forceinline__ v8f wmma4(v2f a, v2f b, v8f c) {
  return __builtin_amdgcn_wmma_f32_16x16x4_f32(false, a, false, b, (short)0, c,
                                               false, false);
}

// ---------------------------------------------------------------------------
// TDM: DMA a 2D f32 tile [Cch rows x CHUNK cols] (row stride Npix elems) from
// global into LDS, padding 1 dword after every 64 dwords so the LDS row
// stride becomes GROW=65 floats (bank-conflict-free fragment reads).
// D# layout per CDNA5 ISA (groups 2/3 unused=0).
// This toolchain declares the 6-arg builtin:
//   (uint32x4 g0, int32x8 g1, int32x4, int32x4, int32x8, i32 cpol)
// ---------------------------------------------------------------------------
__device__ __forceinline__ void tdm_load_chunk(uint32_t lds_byte_off,
                                               const float* gsrc) {
  const uint64_t ga = (uint64_t)(uintptr_t)gsrc;
  v4u g0;
  g0[0] = 1u;                                   // count=1, user mode, no gather
  g0[1] = lds_byte_off;                         // lds_addr [63:32]
  g0[2] = (uint32_t)ga;                         // global_addr low
  g0[3] = (uint32_t)((ga >> 32) & 0x01FFFFFFu)  // global_addr[56:32]
          | (2u << 30);                         // type=2 ("image") [127:126]
  v8i g1;
  g1[0] = (int)((2u << 16)    // data_size=2 -> 4 bytes
                | (1u << 20)  // pad_enable
                | (5u << 22)  // pad_interval: every 64 dwords
                | (0u << 25)); // pad_amount: 1 dword
  g1[1] = (int)((Npix & 0xFFFFu) << 16);            // tensor_dim0[15:0] @ b63:48
  g1[2] = (int)(((Npix >> 16) & 0xFFFFu)            // tensor_dim0[31:16]
                | ((Cch & 0xFFFFu) << 16));         // tensor_dim1[15:0]
  g1[3] = (int)(((Cch >> 16) & 0xFFFFu)             // tensor_dim1[31:16]
                | ((CHUNK & 0xFFFFu) << 16));       // tile_dim0
  g1[4] = (int)(Cch & 0xFFFFu);                     // tile_dim1 (tile_dim2=0)
  g1[5] = (int)Npix;                                // tensor_dim0_stride low
  g1[6] = 0;                                        // stride hi / dim1_stride lo
  g1[7] = 0;                                        // tensor_dim1_stride hi
  const v4i z4 = {0, 0, 0, 0};
  const v8i z8 = {0, 0, 0, 0, 0, 0, 0, 0};
  __builtin_amdgcn_tensor_load_to_lds(g0, g1, z4, z4, z8, 0);
}

__device__ __forceinline__ uint32_t lds_off(const void* p) {
  return (uint32_t)(uintptr_t)p;                // low 32 bits = LDS byte offset
}

// ---------------------------------------------------------------------------
// Kernel 1: f/g/h = W @ x + bias   (per-batch 64x64 @ 64x4096 GEMMs)
// ---------------------------------------------------------------------------
__global__ __launch_bounds__(256) void proj_kernel(
    const float* __restrict__ x,
    const float* __restrict__ Wf, const float* __restrict__ bf,
    const float* __restrict__ Wg, const float* __restrict__ bg,
    const float* __restrict__ Wh, const float* __restrict__ bh,
    float* __restrict__ fOut, float* __restrict__ gOut, float* __restrict__ hOut) {
  const int lane = threadIdx.x & 31, half = lane >> 4, l16 = lane & 15;
  const int t  = blockIdx.x * 8 + (threadIdx.x >> 5);
  const int b  = t >> 8;
  const int n0 = (t & 255) * 16;
  const float* xb = x + b * Cch * Npix;

  v8f accF[4], accG[4], accH[4];
#pragma unroll
  for (int m = 0; m < 4; ++m)
#pragma unroll
    for (int v = 0; v < 8; ++v) {
      const int row = m * 16 + v + 8 * half;   // C/D layout: M = v + 8*half
      accF[m][v] = bf[row];
      accG[m][v] = bg[row];
      accH[m][v] = bh[row];
    }

#pragma unroll 4
  for (int k0 = 0; k0 < Cch; k0 += 4) {
    const int kr = k0 + 2 * half;
    v2f bx;
    bx.x = xb[(kr    ) * Npix + n0 + l16];
    bx.y = xb[(kr + 1) * Npix + n0 + l16];
#pragma unroll
    for (int m = 0; m < 4; ++m) {
      const int wrow = (m * 16 + l16) * Cch + kr;
      v2f af = *reinterpret_cast<const v2f*>(Wf + wrow);
      v2f ag = *reinterpret_cast<const v2f*>(Wg + wrow);
      v2f ah = *reinterpret_cast<const v2f*>(Wh + wrow);
      accF[m] = wmma4(af, bx, accF[m]);
      accG[m] = wmma4(ag, bx, accG[m]);
      accH[m] = wmma4(ah, bx, accH[m]);
    }
  }

  float* fb = fOut + b * Cch * Npix;
  float* gb = gOut + b * Cch * Npix;
  float* hb = hOut + b * Cch * Npix;
#pragma unroll
  for (int m = 0; m < 4; ++m)
#pragma unroll
    for (int v = 0; v < 8; ++v) {
      const int idx = (m * 16 + v + 8 * half) * Npix + n0 + l16;
      fb[idx] = accF[m][v];
      gb[idx] = accG[m][v];
      hb[idx] = accH[m][v];
    }
}

// ---------------------------------------------------------------------------
// Kernel 2: L[b,i] = log( sum_j exp( sum_c f[c,i] g[c,j] ) )
// All 8 waves of a block share batch b; g[b] chunks are TDM-staged into LDS
// (double-buffered) once per block instead of 8x from L2.
// ---------------------------------------------------------------------------
__global__ __launch_bounds__(256) void lse_kernel(
    const float* __restrict__ fbuf, const float* __restrict__ gbuf,
    float* __restrict__ lse) {
  __shared__ float gs[2][Cch * GROW];
  const int lane = threadIdx.x & 31, half = lane >> 4, l16 = lane & 15;
  const int wave = threadIdx.x >> 5;
  const int t  = blockIdx.x * 8 + wave;
  const int b  = t >> 8;                       // uniform across the block
  const int i0 = (t & 255) * 16;
  const float* fb = fbuf + b * Cch * Npix;
  const float* gb = gbuf + b * Cch * Npix;

  v2f af[16];                      // A[M=i][K=c] = f[c][i]; invariant over j
#pragma unroll
  for (int k = 0; k < 16; ++k) {
    const int kr = 4 * k + 2 * half;
    af[k].x = fb[(kr    ) * Npix + i0 + l16];
    af[k].y = fb[(kr + 1) * Npix + i0 + l16];
  }

  float Z[8];
#pragma unroll
  for (int v = 0; v < 8; ++v) Z[v] = 0.0f;

  if (wave == 0) {                             // stage chunk 0
    tdm_load_chunk(lds_off(&gs[0][0]), gb);
    __builtin_amdgcn_s_wait_tensorcnt(0);
  }
  __syncthreads();

  int cur = 0;
  for (int ch = 0; ch < NCH; ++ch) {
    if (wave == 0 && ch + 1 < NCH)             // prefetch next chunk
      tdm_load_chunk(lds_off(&gs[cur ^ 1][0]), gb + (ch + 1) * CHUNK);

    const float* gl = &gs[cur][0];
#pragma unroll
    for (int jt = 0; jt < CHUNK / 16; ++jt) {
      const int j0 = jt * 16;
      v8f s = {0.f, 0.f, 0.f, 0.f, 0.f, 0.f, 0.f, 0.f};
#pragma unroll
      for (int k = 0; k < 16; ++k) {
        const int kr = 4 * k + 2 * half;
        v2f bg_;
        bg_.x = gl[(kr    ) * GROW + j0 + l16];
        bg_.y = gl[(kr + 1) * GROW + j0 + l16];
        s = wmma4(af[k], bg_, s);
      }
#pragma unroll
      for (int v = 0; v < 8; ++v) Z[v] += __expf(s[v]);
    }

    if (wave == 0 && ch + 1 < NCH) __builtin_amdgcn_s_wait_tensorcnt(0);
    __syncthreads();
    cur ^= 1;
  }

#pragma unroll
  for (int v = 0; v < 8; ++v) {
    float z = Z[v];
#pragma unroll
    for (int m = 1; m < 16; m <<= 1) z += __shfl_xor(z, m, 32);
    Z[v] = z;
  }
  if (l16 == 0) {                  // lane 0 -> rows 0..7, lane 16 -> rows 8..15
#pragma unroll
    for (int v = 0; v < 8; ++v)
      lse[b * Npix + i0 + v + 8 * half] = __logf(Z[v]);
  }
}

// ---------------------------------------------------------------------------
// Kernel 3: out[c,j] = gamma * sum_i h[c,i] * exp(s[i,j] - L[i]) + x[c,j]
// f[b] and h[b] i-chunks are TDM-staged (double-buffered); per-wave e-tile
// goes through padded LDS scratch to re-fragment D-layout -> B-layout.
// ---------------------------------------------------------------------------
__global__ __launch_bounds__(256) void attn_out_kernel(
    const float* __restrict__ x, const float* __restrict__ fbuf,
    const float* __restrict__ gbuf, const float* __restrict__ hbuf,
    const float* __restrict__ lse, const float* __restrict__ gamma,
    float* __restrict__ out) {
  __shared__ float fs[2][Cch * GROW];
  __shared__ float hs[2][Cch * GROW];
  __shared__ float etile[8][16 * 17];   // per-wave 16x16 tile, stride 17 pad
  const int lane = threadIdx.x & 31, half = lane >> 4, l16 = lane & 15;
  const int wave = threadIdx.x >> 5;
  const int t  = blockIdx.x * 8 + wave;
  const int b  = t >> 8;                       // uniform across the block
  const int j0 = (t & 255) * 16;
  const float* fb = fbuf + b * Cch * Npix;
  const float* gb = gbuf + b * Cch * Npix;
  const float* hb = hbuf + b * Cch * Npix;
  const float* Ls = lse + b * Npix;
  float* my = &etile[wave][0];

  v2f bg_[16];                     // B[K=c][N=j] = g[c][j]; invariant over i
#pragma unroll
  for (int k = 0; k < 16; ++k) {
    const int kr = 4 * k + 2 * half;
    bg_[k].x = gb[(kr    ) * Npix + j0 + l16];
    bg_[k].y = gb[(kr + 1) * Npix + j0 + l16];
  }

  v8f acc[4];
#pragma unroll
  for (int m = 0; m < 4; ++m)
#pragma unroll
    for (int v = 0; v < 8; ++v) acc[m][v] = 0.0f;

  if (wave == 0) {                             // stage chunk 0 of f and h
    tdm_load_chunk(lds_off(&fs[0][0]), fb);
    tdm_load_chunk(lds_off(&hs[0][0]), hb);
    __builtin_amdgcn_s_wait_tensorcnt(0);
  }
  __syncthreads();

  int cur = 0;
  for (int ch = 0; ch < NCH; ++ch) {
    if (wave == 0 && ch + 1 < NCH) {           // prefetch next chunks
      tdm_load_chunk(lds_off(&fs[cur ^ 1][0]), fb + (ch + 1) * CHUNK);
      tdm_load_chunk(lds_off(&hs[cur ^ 1][0]), hb + (ch + 1) * CHUNK);
    }
    const float* fl = &fs[cur][0];
    const float* hl = &hs[cur][0];

#pragma unroll
    for (int itl = 0; itl < CHUNK / 16; ++itl) {
      const int il0 = itl * 16;                // i offset within chunk
      const int i0  = ch * CHUNK + il0;        // global i offset
      // score tile: A[M=i][K=c] = f[c][i] (from LDS)
      v8f s = {0.f, 0.f, 0.f, 0.f, 0.f, 0.f, 0.f, 0.f};
#pragma unroll
      for (int k = 0; k < 16; ++k) {
        const int kr = 4 * k + 2 * half;
        v2f af;
        af.x = fl[(kr    ) * GROW + il0 + l16];
        af.y = fl[(kr + 1) * GROW + il0 + l16];
        s = wmma4(af, bg_[k], s);
      }
      // e = exp(s - L[i]); write row-major into this wave's LDS tile
#pragma unroll
      for (int v = 0; v < 8; ++v) {
        const int irow = v + 8 * half;                    // D layout row
        const float e = __expf(s[v] - Ls[i0 + irow]);
        my[irow * 17 + l16] = e;
      }
      // out accumulation: A[M=c][K=i] = h[c][i] (from LDS),
      // B[K=i][N=j] = e tile re-read from LDS in B-fragment layout.
#pragma unroll
      for (int ks = 0; ks < 4; ++ks) {
        const int kr = 4 * ks + 2 * half;
        v2f eb;
        eb.x = my[(kr    ) * 17 + l16];
        eb.y = my[(kr + 1) * 17 + l16];
#pragma unroll
        for (int m = 0; m < 4; ++m) {
          v2f ah;
          const int hbase = (m * 16 + l16) * GROW + il0 + kr;
          ah.x = hl[hbase];
          ah.y = hl[hbase + 1];
          acc[m] = wmma4(ah, eb, acc[m]);
        }
      }
    }

    if (wave == 0 && ch + 1 < NCH) __builtin_amdgcn_s_wait_tensorcnt(0);
    __syncthreads();
    cur ^= 1;
  }

  const float gm = gamma[0];
  const float* xb = x + b * Cch * Npix;
  float* ob = out + b * Cch * Npix;
#pragma unroll
  for (int m = 0; m < 4; ++m)
#pragma unroll
    for (int v = 0; v < 8; ++v) {
      const int idx = (m * 16 + v + 8 * half) * Npix + j0 + l16;
      ob[idx] = gm * acc[m][v] + xb[idx];
    }
}

extern "C" void kernel_launch(void* const* d_in, const int* in_sizes, int n_in,
                              void* d_out, int out_size, void* d_ws,
                              size_t ws_size, hipStream_t stream) {
  const float* x     = (const float*)d_in[0];
  const float* Wf    = (const float*)d_in[1];
  const float* bf    = (const float*)d_in[2];
  const float* Wg    = (const float*)d_in[3];
  const float* bg    = (const float*)d_in[4];
  const float* Wh    = (const float*)d_in[5];
  const float* bh    = (const float*)d_in[6];
  const float* gamma = (const float*)d_in[7];
  float* out = (float*)d_out;

  float* fbuf = (float*)d_ws;          // [B,C,N]
  float* gbuf = fbuf + BCN;            // [B,C,N]
  float* hbuf = gbuf + BCN;            // [B,C,N]
  float* lseb = hbuf + BCN;            // [B,N] log-sum-exp per score row

  dim3 grd(256), blk(256);             // 8 waves/block, 2048 tile-waves total
  hipLaunchKernelGGL(proj_kernel, grd, blk, 0, stream, x, Wf, bf, Wg, bg, Wh,
                     bh, fbuf, gbuf, hbuf);
  hipLaunchKernelGGL(lse_kernel, grd, blk, 0, stream, fbuf, gbuf, lseb);
  hipLaunchKernelGGL(attn_out_kernel, grd, blk, 0, stream, x, fbuf, gbuf, hbuf,
                     lseb, gamma, out);
}